// OverlappingTrianglesLoss_17626545783308
// MI455X (gfx1250) — compile-verified
//
#include <hip/hip_runtime.h>
#include <math.h>

// ---------------------------------------------------------------------------
// OverlappingTrianglesLoss for MI455X (gfx1250, wave32).
// Distance matrix tiles computed with V_WMMA_F32_16X16X4_F32 (fp32, K=4):
//   A (16x4)  : per-centroid rows  (-2cx, -2cy, -2cz, cc)
//   B (4x16)  : per-point  columns ( px,   py,   pz,  1 )
//   C (16x16) : pp broadcast per point column (free add via accumulator)
//   D = A*B+C : pp + cc - 2*dot(c,p)  ==  d2   (power-of-2-exact scaling,
//   matches the reference's  pp + cc - 2*(P @ C^T)  formulation)
// One-deep software prefetch of the A tile hides the L2 load latency under
// the ~100-VALU-op top-5 insertion network. Deterministic reductions.
// ---------------------------------------------------------------------------

typedef __attribute__((ext_vector_type(2))) float v2f;
typedef __attribute__((ext_vector_type(8))) float v8f;

#define NV_   4000
#define NF_   6000
#define NS_   10
#define NP_   (NF_ * NS_)          // 60000 sampled points
#define KSEL  5
#define EPSA  1e-6f

#define WPB   4                    // waves per block (wave32)
#define TPB   (WPB * 32)           // 128 threads
#define NTILE (NF_ / 16)           // 375 centroid tiles (6000 % 16 == 0)
#define NWAVE_TOT (NP_ / 16)       // 3750 waves (60000 % 16 == 0)
#define NBLK  ((NWAVE_TOT + WPB - 1) / WPB)   // 938 blocks

struct F3 { float x, y, z; };
__device__ __forceinline__ F3 f3sub(F3 a, F3 b) { return {a.x - b.x, a.y - b.y, a.z - b.z}; }
__device__ __forceinline__ F3 f3cross(F3 a, F3 b) {
  return {a.y * b.z - a.z * b.y, a.z * b.x - a.x * b.z, a.x * b.y - a.y * b.x};
}
__device__ __forceinline__ float f3dot(F3 a, F3 b) { return a.x * b.x + a.y * b.y + a.z * b.z; }
__device__ __forceinline__ F3 ldv(const float* __restrict__ verts, long long i) {
  return {verts[i * 3 + 0], verts[i * 3 + 1], verts[i * 3 + 2]};
}

// ---------------------------------------------------------------------------
// Kernel 1: centroids, packed for the WMMA A-operand: (-2cx,-2cy,-2cz, cc)
// ---------------------------------------------------------------------------
__global__ void centroid_kernel(const float* __restrict__ verts,
                                const long long* __restrict__ faces,
                                float4* __restrict__ cent4) {
  int f = blockIdx.x * blockDim.x + threadIdx.x;
  if (f >= NF_) return;
  long long i0 = faces[3 * f + 0], i1 = faces[3 * f + 1], i2 = faces[3 * f + 2];
  F3 a = ldv(verts, i0), b = ldv(verts, i1), c = ldv(verts, i2);
  float cx = (a.x + b.x + c.x) / 3.0f;
  float cy = (a.y + b.y + c.y) / 3.0f;
  float cz = (a.z + b.z + c.z) / 3.0f;
  float cc = cx * cx + cy * cy + cz * cz;
  cent4[f] = make_float4(-2.0f * cx, -2.0f * cy, -2.0f * cz, cc);
}

// ---------------------------------------------------------------------------
// Kernel 2: fused point-gen -> WMMA distance tiles -> top-5 -> inside test
// Each wave owns 16 points; lanes L and L+16 hold the same point (B/C/D
// column striping), each covering 8 of the 16 centroid rows per tile.
// ---------------------------------------------------------------------------
__global__ void __launch_bounds__(TPB)
knn_kernel(const float* __restrict__ verts,
           const long long* __restrict__ faces,
           const float* __restrict__ uarr,
           const float* __restrict__ varr,
           const float2* __restrict__ cent2,     // 2x float2 per face
           float* __restrict__ partials) {
  __shared__ float sd[WPB][32][KSEL];
  __shared__ int   si[WPB][32][KSEL];
  __shared__ float ssum[WPB][16];

  const int lane  = threadIdx.x & 31;
  const int wave  = threadIdx.x >> 5;
  const int half  = lane >> 4;     // 0: K rows 0..1 / M rows 0..7 ; 1: K 2..3 / M 8..15
  const int sub   = lane & 15;     // point column within the wave's 16-point group
  const int gwave = blockIdx.x * WPB + wave;
  const int praw  = gwave * 16 + sub;
  const bool pvalid = praw < NP_;
  const int pidx  = pvalid ? praw : (NP_ - 1);   // clamp tail waves, zeroed later

  if (lane < 16) ssum[wave][sub] = 0.0f;

  // ---- generate my sampled point (both half-lanes of a column compute it) --
  const int fp = pidx / NS_;
  float uu = uarr[pidx], vv = varr[pidx];
  if (uu + vv > 1.0f) { uu = 1.0f - uu; vv = 1.0f - vv; }
  long long j0 = faces[3 * fp + 0], j1 = faces[3 * fp + 1], j2 = faces[3 * fp + 2];
  F3 q0 = ldv(verts, j0), q1 = ldv(verts, j1), q2 = ldv(verts, j2);
  const float w0 = 1.0f - uu - vv;
  F3 P = { q0.x * w0 + q1.x * uu + q2.x * vv,
           q0.y * w0 + q1.y * uu + q2.y * vv,
           q0.z * w0 + q1.z * uu + q2.z * vv };
  const float pp = P.x * P.x + P.y * P.y + P.z * P.z;

  // B operand (4x16): column = point; lanes 0-15 supply K0,K1; 16-31 K2,K3.
  v2f Bf;
  Bf.x = half ? P.z : P.x;
  Bf.y = half ? 1.0f : P.y;

  // C operand: pp broadcast over this lane's 8 rows (all same point column),
  // so D = pp + cc - 2*dot  ==  d2 with no extra VALU adds.
  v8f Cpp;
#pragma unroll
  for (int j = 0; j < 8; ++j) Cpp[j] = pp;

  // ---- per-lane top-5 (ascending d2; ties keep lowest index: indices are
  //      visited in ascending order and insertion uses strict '<') ----------
  float bd[KSEL];
  int   bi[KSEL];
#pragma unroll
  for (int j = 0; j < KSEL; ++j) { bd[j] = 3.4e38f; bi[j] = 0x7FFFFFFF; }

  // Lane-invariant A-tile addressing: one float2 per lane, 32 float2 / tile.
  const float2* aptr = cent2 + (sub * 2 + half);
  float2 ca = aptr[0];                         // prefetch tile 0

  for (int t = 0; t < NTILE; ++t) {
    v2f Af; Af.x = ca.x; Af.y = ca.y;

    // Prefetch next tile before the long insertion stretch hides its latency.
    const int tn = (t + 1 < NTILE) ? (t + 1) : t;
    float2 ca_next = aptr[tn * 32];

    // D[m][n] = pp_n + cc_m - 2*dot(c_m, p_n)
    v8f D = __builtin_amdgcn_wmma_f32_16x16x4_f32(
        /*neg_a=*/false, Af, /*neg_b=*/false, Bf,
        /*c_mod=*/(short)0, Cpp, /*reuse_a=*/false, /*reuse_b=*/false);

    const int mbase = t * 16 + half * 8;   // C/D rows: half 0 -> M 0..7, half 1 -> M 8..15
#pragma unroll
    for (int r = 0; r < 8; ++r) {
      const float d2 = D[r];
      const int fidx = mbase + r;
      if (d2 < bd[4]) {
        const bool c0 = d2 < bd[0], c1 = d2 < bd[1], c2 = d2 < bd[2], c3 = d2 < bd[3];
        bd[4] = c3 ? bd[3] : d2;                bi[4] = c3 ? bi[3] : fidx;
        bd[3] = c2 ? bd[2] : (c3 ? d2 : bd[3]); bi[3] = c2 ? bi[2] : (c3 ? fidx : bi[3]);
        bd[2] = c1 ? bd[1] : (c2 ? d2 : bd[2]); bi[2] = c1 ? bi[1] : (c2 ? fidx : bi[2]);
        bd[1] = c0 ? bd[0] : (c1 ? d2 : bd[1]); bi[1] = c0 ? bi[0] : (c1 ? fidx : bi[1]);
        bd[0] = c0 ? d2 : bd[0];                bi[0] = c0 ? fidx : bi[0];
      }
    }
    ca = ca_next;
  }

  // ---- merge the two half-lists per point through LDS ---------------------
#pragma unroll
  for (int j = 0; j < KSEL; ++j) { sd[wave][lane][j] = bd[j]; si[wave][lane][j] = bi[j]; }
  __syncthreads();

  if (lane < 16) {
    float lsum = 0.0f;
    int ia = 0, ib = 0;
#pragma unroll
    for (int k = 0; k < KSEL; ++k) {
      float da = sd[wave][sub][ia];       int xa = si[wave][sub][ia];
      float db = sd[wave][sub + 16][ib];  int xb = si[wave][sub + 16][ib];
      bool ta = (da < db) || (da == db && xa < xb);   // (d2, idx) lexicographic
      int fidx = ta ? xa : xb;
      if (ta) ++ia; else ++ib;

      // inside-test against face fidx
      long long k0 = faces[3 * fidx + 0], k1 = faces[3 * fidx + 1], k2 = faces[3 * fidx + 2];
      F3 A3 = ldv(verts, k0), B3 = ldv(verts, k1), C3 = ldv(verts, k2);
      F3 nn = f3cross(f3sub(B3, A3), f3sub(C3, A3));
      float area = 0.5f * sqrtf(f3dot(nn, nn));
      float uc = f3dot(f3cross(f3sub(C3, B3), f3sub(P, B3)), nn);
      float vc = f3dot(f3cross(f3sub(A3, C3), f3sub(P, C3)), nn);
      float wc = f3dot(f3cross(f3sub(B3, A3), f3sub(P, A3)), nn);
      bool inside = (fabsf(area) > EPSA) && (uc >= 0.0f) && (vc >= 0.0f) && (wc >= 0.0f);
      if (inside) lsum += area;
    }
    if (pvalid) ssum[wave][sub] = lsum;
  }
  __syncthreads();

  if (threadIdx.x == 0) {
    float s = 0.0f;
    for (int w = 0; w < WPB; ++w)
      for (int i = 0; i < 16; ++i) s += ssum[w][i];
    partials[blockIdx.x] = s;
  }
}

// ---------------------------------------------------------------------------
// Kernel 3: deterministic final reduction of block partials
// ---------------------------------------------------------------------------
__global__ void reduce_kernel(const float* __restrict__ partials, int n,
                              float* __restrict__ out) {
  if (blockIdx.x == 0 && threadIdx.x == 0) {
    float s = 0.0f;
    for (int i = 0; i < n; ++i) s += partials[i];
    out[0] = s;
  }
}

// ---------------------------------------------------------------------------
extern "C" void kernel_launch(void* const* d_in, const int* in_sizes, int n_in,
                              void* d_out, int out_size, void* d_ws, size_t ws_size,
                              hipStream_t stream) {
  (void)in_sizes; (void)n_in; (void)out_size; (void)ws_size;
  const float*      verts = (const float*)d_in[0];       // (4000,3) f32
  const long long*  faces = (const long long*)d_in[1];   // (6000,3) i64
  const float*      u     = (const float*)d_in[2];       // (6000,10,1) f32
  const float*      v     = (const float*)d_in[3];       // (6000,10,1) f32
  float*            out   = (float*)d_out;               // scalar f32

  char* ws = (char*)d_ws;
  float4* cent4 = (float4*)ws;                                    // 96000 B
  size_t cent_bytes = ((size_t)NF_ * sizeof(float4) + 255) & ~(size_t)255;
  float* partials = (float*)(ws + cent_bytes);                    // NBLK * 4 B

  centroid_kernel<<<(NF_ + 255) / 256, 256, 0, stream>>>(verts, faces, cent4);
  knn_kernel<<<NBLK, TPB, 0, stream>>>(verts, faces, u, v,
                                       (const float2*)cent4, partials);
  reduce_kernel<<<1, 32, 0, stream>>>(partials, NBLK, out);
}